// MapGuide_24489903522008
// MI455X (gfx1250) — compile-verified
//
#include <hip/hip_runtime.h>
#include <math.h>

// MapGuide: per-batch sequential guided-gradient scan.
// B=64 independent chains; each chain = 112 timesteps x 4 dependent gather
// rounds (gx/gy at the same data-dependent [i0,i1]). Latency-bound pointer
// chase -> 1 lane per batch, wave32-native (2 waves), overlapping paired
// gathers, and global_prefetch_b8 of the predicted future gather addresses
// (base coord of t+PF plus current offset) to stage lines in L2/L0 ahead of
// the dependency chain. 128MB of gradient maps fits MI455X's 192MB L2.
__global__ __launch_bounds__(32) void MapGuide_kernel(
    const float* __restrict__ x,         // [B,T,2]
    const float* __restrict__ gx,        // [B,H,W]
    const float* __restrict__ gy,        // [B,H,W]
    const float* __restrict__ center,    // [2]
    const float* __restrict__ std_scale, // [2]
    const int*  __restrict__ obs_len_p,  // scalar
    const int*  __restrict__ n_guide_p,  // scalar
    float* __restrict__ out,             // [B,T,2]
    int B, int T, int W, long long HW)
{
    const int b = blockIdx.x * 32 + threadIdx.x;
    if (b >= B) return;

    const int obs_len = *obs_len_p;
    const int n_guide = *n_guide_p;
    const int maxc    = W - 1;                 // reference clamps both dims to W-1

    const float cx = center[0],    cy = center[1];
    const float sx = std_scale[0], sy = std_scale[1];

    const float* __restrict__ gxb  = gx + (long long)b * HW;
    const float* __restrict__ gyb  = gy + (long long)b * HW;
    const float* __restrict__ xb   = x  + (size_t)b * (size_t)T * 2;
    float*       __restrict__ outb = out + (size_t)b * (size_t)T * 2;

    // Rows t < obs_len are exact zeros (offsets start at 0).
    for (int t = 0; t < obs_len && t < T; ++t) {
        outb[2 * t]     = 0.0f;
        outb[2 * t + 1] = 0.0f;
    }

    float off0 = 0.0f, off1 = 0.0f;
    const int PF = 3;   // prefetch distance (timesteps ahead of the chain)

    for (int t = obs_len; t < T; ++t) {
        // ---- prefetch predicted gather line for step t+PF (hint only;
        //      offset drift within PF steps is tiny since |grad| ~ 0.5) ----
        if (t + PF < T) {
            float q0 = xb[2 * (t + PF)]     * sx + cx + off0;
            float q1 = xb[2 * (t + PF) + 1] * sy + cy + off1;
            int j0 = min(max((int)q0, 0), maxc);
            int j1 = min(max((int)q1, 0), maxc);
            const size_t pidx = (size_t)j0 * (size_t)W + (size_t)j1;
            __builtin_prefetch(&gxb[pidx], 0, 3);   // -> global_prefetch_b8
            __builtin_prefetch(&gyb[pidx], 0, 3);
        }

        // ---- base coordinate for this timestep ----
        const float c0 = xb[2 * t]     * sx + cx;
        const float c1 = xb[2 * t + 1] * sy + cy;

        // clamped = clip(coord + offset, 0, maxc).astype(int32)
        float p0 = c0 + off0;
        float p1 = c1 + off1;
        int i0 = (int)fminf(fmaxf(p0, 0.0f), (float)maxc);
        int i1 = (int)fminf(fmaxf(p1, 0.0f), (float)maxc);

        // ---- dependent guide loop: two overlapped gathers per round ----
        float g0 = 0.0f, g1 = 0.0f;
        for (int k = 0; k < n_guide; ++k) {
            const size_t idx = (size_t)i0 * (size_t)W + (size_t)i1;
            const float gv0 = gxb[idx];   // issued back-to-back: latencies
            const float gv1 = gyb[idx];   // overlap under one s_wait_loadcnt
            g0 -= gv0;
            g1 -= gv1;
            i0 = min(max(i0 - (int)gv0, 0), maxc);  // (int) truncates toward 0,
            i1 = min(max(i1 - (int)gv1, 0), maxc);  // matching astype(int32)
        }

        off0 += g0;
        off1 += g1;

        // grad = offset / std_scale
        outb[2 * t]     = off0 / sx;
        outb[2 * t + 1] = off1 / sy;
    }
}

extern "C" void kernel_launch(void* const* d_in, const int* in_sizes, int n_in,
                              void* d_out, int out_size, void* d_ws, size_t ws_size,
                              hipStream_t stream) {
    // setup_inputs order:
    //   0: x [B,T,2] f32     1: cond0 [B,2] f32 (dead: only writes row 0, scan
    //      starts at obs_len and rows < obs_len output exact zeros)
    //   2: grad_x [B,H,W]    3: grad_y [B,H,W]
    //   4: center [2]        5: std_scale [2]
    //   6: obs_len (int)     7: n_guide_steps (int)
    (void)d_ws; (void)ws_size; (void)n_in;

    const float* x      = (const float*)d_in[0];
    const float* gx     = (const float*)d_in[2];
    const float* gy     = (const float*)d_in[3];
    const float* center = (const float*)d_in[4];
    const float* stds   = (const float*)d_in[5];
    const int*   obs_p  = (const int*)d_in[6];
    const int*   ngd_p  = (const int*)d_in[7];
    float*       out    = (float*)d_out;

    const int B = in_sizes[1] / 2;                        // cond0 = [B,2]
    const int T = in_sizes[0] / (2 * B);                  // x = [B,T,2]
    const long long HW = (long long)in_sizes[2] / B;      // grad_x = [B,H,W]
    const int W = (int)llround(sqrt((double)HW));         // square maps (512)

    const int threads = 32;                               // one wave32 per block
    const int blocks  = (B + threads - 1) / threads;      // 64 batches -> 2 waves
    MapGuide_kernel<<<blocks, threads, 0, stream>>>(
        x, gx, gy, center, stds, obs_p, ngd_p, out, B, T, W, HW);
}